// S4Layer_68427418960360
// MI455X (gfx1250) — compile-verified
//
#include <hip/hip_runtime.h>
#include <math.h>

namespace {

constexpr int   LFULL  = 16384;
constexpr int   LSIDE  = 128;          // 16384 = 128 x 128 four-step FFT
constexpr int   NSTATE = 512;
constexpr int   BATCHN = 256;
constexpr float STEPF  = 0.001f;
constexpr float TWO_PI = 6.28318530717958647692f;

constexpr int   VP     = 136;          // LDS pitch for V: transposed reads hit all 64 banks
constexpr int   GP     = 40;           // LDS pitch for G panel: B-frag reads conflict-free
constexpr int   PANEL  = 32;           // G panel width (columns)
constexpr size_t LDS_BYTES = (size_t)(2*LSIDE*VP + 2*LSIDE*GP) * sizeof(float); // 180224 B

typedef __attribute__((ext_vector_type(2))) float v2f;
typedef __attribute__((ext_vector_type(8))) float v8f;

#define V8ZERO {0.f,0.f,0.f,0.f,0.f,0.f,0.f,0.f}

__device__ __forceinline__ v8f wmma4(v2f a, v2f b, v8f c) {
  // D = A(16x4 f32) * B(4x16 f32) + C(16x16 f32), full fp32 matrix pipe
  return __builtin_amdgcn_wmma_f32_16x16x4_f32(
      false, a, false, b, (short)0, c, false, false);
}

struct cplx { float r, i; };
__device__ __forceinline__ cplx cmul(cplx a, cplx b) {
  return { a.r*b.r - a.i*b.i, a.r*b.i + a.i*b.r };
}
__device__ __forceinline__ cplx cdiv(cplx a, cplx b) {
  float id = 1.f / (b.r*b.r + b.i*b.i);
  return { (a.r*b.r + a.i*b.i)*id, (a.i*b.r - a.r*b.i)*id };
}

// ---------------------------------------------------------------------------
// Setup: DFT-128 matrix M[a][b]=W128^(ab) and four-step twiddle T[a][b]=W16384^(ab)
// ---------------------------------------------------------------------------
__global__ void __launch_bounds__(256) k_setup(float* __restrict__ Mre,
                                               float* __restrict__ Mim,
                                               float* __restrict__ Tre,
                                               float* __restrict__ Tim) {
  int i = blockIdx.x * blockDim.x + threadIdx.x;      // 0..16383
  int a = i >> 7, b = i & 127;
  int p = a * b;
  float s, c;
  sincosf(-TWO_PI * (float)(p & 127)   * (1.f/128.f),   &s, &c);
  Mre[i] = c; Mim[i] = s;
  sincosf(-TWO_PI * (float)(p & 16383) * (1.f/16384.f), &s, &c);
  Tre[i] = c; Tim[i] = s;
}

// ---------------------------------------------------------------------------
// DPLR Cauchy kernel + Woodbury: one thread per frequency l
// ---------------------------------------------------------------------------
__global__ void __launch_bounds__(256) k_cauchy(
    const float* __restrict__ Lr, const float* __restrict__ Li,
    const float* __restrict__ Pr, const float* __restrict__ Pi,
    const float* __restrict__ Br, const float* __restrict__ Bi,
    const float* __restrict__ Cr, const float* __restrict__ Ci,
    float* __restrict__ Kre, float* __restrict__ Kim) {
  int l = blockIdx.x * blockDim.x + threadIdx.x;
  float so, co;
  sincosf(-TWO_PI * (float)l * (1.f/(float)LFULL), &so, &co);
  cplx onep{1.f + co,  so};
  cplx onem{1.f - co, -so};
  cplx g  = cdiv({(2.f/STEPF)*onem.r, (2.f/STEPF)*onem.i}, onep);
  cplx cc = cdiv({2.f, 0.f}, onep);
  cplx k00{0,0}, k01{0,0}, k10{0,0}, k11{0,0};
  for (int n = 0; n < NSTATE; ++n) {
    float lr = Lr[n], li = Li[n];
    float pr = Pr[n], pi = Pi[n];
    float br = Br[n], bi = Bi[n];
    float cr = Cr[n], ci = Ci[n];
    float dr = g.r - lr, di = g.i - li;
    float id = 1.f / (dr*dr + di*di);
    cplx inv{ dr*id, -di*id };
    cplx a0{ cr, -ci };           // conj(C)
    cplx a1{ pr, -pi };           // conj(P)
    cplx b0{ br,  bi };
    cplx b1{ pr,  pi };
    cplx t;
    t = cmul(cmul(a0, b0), inv); k00.r += t.r; k00.i += t.i;
    t = cmul(cmul(a0, b1), inv); k01.r += t.r; k01.i += t.i;
    t = cmul(cmul(a1, b0), inv); k10.r += t.r; k10.i += t.i;
    t = cmul(cmul(a1, b1), inv); k11.r += t.r; k11.i += t.i;
  }
  cplx w = cdiv(cmul(k01, k10), {1.f + k11.r, k11.i});
  cplx K = cmul(cc, {k00.r - w.r, k00.i - w.i});
  Kre[l] = K.r; Kim[l] = K.i;
}

// ---------------------------------------------------------------------------
// Fused per-row pipeline: y = Re( FFT( conj(K .* FFT(x)) ) ) / L
// Both length-16384 FFTs are four-step (128x128) chained GEMMs on the f32
// WMMA pipe; the full 128x128 complex spectrum V lives in LDS (no HBM trip).
// ---------------------------------------------------------------------------
__global__ void __launch_bounds__(256) k_s4(
    const float* __restrict__ x,
    const float* __restrict__ Mre, const float* __restrict__ Mim,
    const float* __restrict__ Tre, const float* __restrict__ Tim,
    const float* __restrict__ Kre, const float* __restrict__ Kim,
    float* __restrict__ out) {
  extern __shared__ float lds[];
  float* VreL = lds;                       // [128][VP]
  float* VimL = VreL + LSIDE * VP;
  float* Gre  = VimL + LSIDE * VP;         // [128][GP] (32-wide panel)
  float* Gim  = Gre  + LSIDE * GP;

  const int b    = blockIdx.x;
  const float* xr = x + (size_t)b * LFULL;
  const int lane = threadIdx.x & 31;
  const int wave = threadIdx.x >> 5;
  const int m    = lane & 15;
  const int hi   = lane >> 4;
  const int tm   = wave;                   // this wave owns tile-row tm
  const int rowA = tm*16 + m;

  // Hoisted A-fragments of M for the outer-DFT stage of BOTH phases.
  // M is symmetric: M[rowA][k] == M[k][rowA] -> coalesced loads, 128 VGPRs.
  v2f mAre[32], mAim[32];
#pragma unroll
  for (int q = 0; q < 32; ++q) {
    const int k0 = q*4 + hi*2;
    mAre[q].x = Mre[(k0 << 7) + rowA];
    mAre[q].y = Mre[((k0+1) << 7) + rowA];
    mAim[q].x = Mim[(k0 << 7) + rowA];
    mAim[q].y = Mim[((k0+1) << 7) + rowA];
  }

  // ---------------- forward: V = conj(K .* FFT(x_row)) -> LDS -------------
  for (int p = 0; p < 4; ++p) {
    // Stage A: F = Y*M (Y real, Y[n1][n2]=x[n1+128*n2]); G = T .* F -> panel
    for (int t = 0; t < 2; ++t) {
      const int tn = p*2 + t;
      v8f accR = V8ZERO, accI = V8ZERO;
      const int coln = tn*16 + m;
      for (int q = 0; q < 32; ++q) {
        const int k0 = q*4 + hi*2;
        v2f a, bre, bim;
        a.x   = xr[rowA + (k0 << 7)];
        a.y   = xr[rowA + ((k0+1) << 7)];
        bre.x = Mre[(k0 << 7) + coln];
        bre.y = Mre[((k0+1) << 7) + coln];
        bim.x = Mim[(k0 << 7) + coln];
        bim.y = Mim[((k0+1) << 7) + coln];
        accR = wmma4(a, bre, accR);
        accI = wmma4(a, bim, accI);
      }
#pragma unroll
      for (int j = 0; j < 8; ++j) {
        const int rr = tm*16 + hi*8 + j;
        const int cc = tn*16 + m;
        const float tr = Tre[(rr << 7) + cc];
        const float ti = Tim[(rr << 7) + cc];
        const float fr = accR[j], fi = accI[j];
        Gre[rr*GP + (cc - p*PANEL)] = fr*tr - fi*ti;
        Gim[rr*GP + (cc - p*PANEL)] = fr*ti + fi*tr;
      }
    }
    __syncthreads();
    // Stage C: U = M*G ; V = conj(K .* U) -> LDS spectrum
    for (int t = 0; t < 2; ++t) {
      const int tn = p*2 + t;
      v8f RR = V8ZERO, II = V8ZERO, RI = V8ZERO, IR = V8ZERO;
      const int ccL = tn*16 + m - p*PANEL;
#pragma unroll
      for (int q = 0; q < 32; ++q) {
        const int k0 = q*4 + hi*2;
        v2f bre, bim;
        bre.x = Gre[k0*GP + ccL];
        bre.y = Gre[(k0+1)*GP + ccL];
        bim.x = Gim[k0*GP + ccL];
        bim.y = Gim[(k0+1)*GP + ccL];
        RR = wmma4(mAre[q], bre, RR);
        II = wmma4(mAim[q], bim, II);
        RI = wmma4(mAre[q], bim, RI);
        IR = wmma4(mAim[q], bre, IR);
      }
#pragma unroll
      for (int j = 0; j < 8; ++j) {
        const int rr = tm*16 + hi*8 + j;
        const int cc = tn*16 + m;
        const int kg = (rr << 7) + cc;          // natural frequency index
        const float ur = RR[j] - II[j];
        const float ui = RI[j] + IR[j];
        const float kr = Kre[kg], ki = Kim[kg];
        VreL[rr*VP + cc] =   kr*ur - ki*ui;
        VimL[rr*VP + cc] = -(kr*ui + ki*ur);    // conj for inverse-as-forward
      }
    }
    __syncthreads();
  }

  // ---------------- inverse: y = Re(FFT(V)) / L ---------------------------
  for (int p = 0; p < 4; ++p) {
    // Stage A: F' = Y'*M with Y'[a][c] = V[a+128c] = Vmat[c][a] (LDS transpose)
    for (int t = 0; t < 2; ++t) {
      const int tn = p*2 + t;
      v8f RR = V8ZERO, II = V8ZERO, RI = V8ZERO, IR = V8ZERO;
      const int coln = tn*16 + m;
      for (int q = 0; q < 32; ++q) {
        const int k0 = q*4 + hi*2;
        v2f ar, ai, bre, bim;
        ar.x  = VreL[k0*VP + rowA];
        ar.y  = VreL[(k0+1)*VP + rowA];
        ai.x  = VimL[k0*VP + rowA];
        ai.y  = VimL[(k0+1)*VP + rowA];
        bre.x = Mre[(k0 << 7) + coln];
        bre.y = Mre[((k0+1) << 7) + coln];
        bim.x = Mim[(k0 << 7) + coln];
        bim.y = Mim[((k0+1) << 7) + coln];
        RR = wmma4(ar, bre, RR);
        II = wmma4(ai, bim, II);
        RI = wmma4(ar, bim, RI);
        IR = wmma4(ai, bre, IR);
      }
#pragma unroll
      for (int j = 0; j < 8; ++j) {
        const int rr = tm*16 + hi*8 + j;
        const int cc = tn*16 + m;
        const float tr = Tre[(rr << 7) + cc];
        const float ti = Tim[(rr << 7) + cc];
        const float fr = RR[j] - II[j];
        const float fi = RI[j] + IR[j];
        Gre[rr*GP + (cc - p*PANEL)] = fr*tr - fi*ti;
        Gim[rr*GP + (cc - p*PANEL)] = fr*ti + fi*tr;
      }
    }
    __syncthreads();
    // Stage C: Ym = M*G ; y = Re(Ym)/L -> global output (real part only)
    for (int t = 0; t < 2; ++t) {
      const int tn = p*2 + t;
      v8f RR = V8ZERO, II = V8ZERO;
      const int ccL = tn*16 + m - p*PANEL;
#pragma unroll
      for (int q = 0; q < 32; ++q) {
        const int k0 = q*4 + hi*2;
        v2f bre, bim;
        bre.x = Gre[k0*GP + ccL];
        bre.y = Gre[(k0+1)*GP + ccL];
        bim.x = Gim[k0*GP + ccL];
        bim.y = Gim[(k0+1)*GP + ccL];
        RR = wmma4(mAre[q], bre, RR);
        II = wmma4(mAim[q], bim, II);
      }
#pragma unroll
      for (int j = 0; j < 8; ++j) {
        const int rr = tm*16 + hi*8 + j;
        const int cc = tn*16 + m;
        out[(size_t)b*LFULL + (rr << 7) + cc] = (RR[j] - II[j]) * (1.f/(float)LFULL);
      }
    }
    __syncthreads();
  }
}

} // anonymous namespace

// ---------------------------------------------------------------------------
extern "C" void kernel_launch(void* const* d_in, const int* in_sizes, int n_in,
                              void* d_out, int out_size, void* d_ws, size_t ws_size,
                              hipStream_t stream) {
  (void)in_sizes; (void)n_in; (void)out_size; (void)ws_size;
  const float* x  = (const float*)d_in[0];
  const float* Lr = (const float*)d_in[1];
  const float* Li = (const float*)d_in[2];
  const float* Pr = (const float*)d_in[3];
  const float* Pi = (const float*)d_in[4];
  const float* Br = (const float*)d_in[5];
  const float* Bi = (const float*)d_in[6];
  const float* Cr = (const float*)d_in[7];
  const float* Ci = (const float*)d_in[8];

  float* ws  = (float*)d_ws;                // only 384 KB of tables now
  float* Mre = ws;
  float* Mim = Mre + LFULL;
  float* Tre = Mim + LFULL;
  float* Tim = Tre + LFULL;
  float* Kre = Tim + LFULL;
  float* Kim = Kre + LFULL;
  float* out = (float*)d_out;

  // Allow >64KB dynamic LDS (WGP has 320KB); harmless if already permitted.
  (void)hipFuncSetAttribute((const void*)k_s4,
                            hipFuncAttributeMaxDynamicSharedMemorySize,
                            (int)LDS_BYTES);

  k_setup <<<dim3(LFULL/256), dim3(256), 0, stream>>>(Mre, Mim, Tre, Tim);
  k_cauchy<<<dim3(LFULL/256), dim3(256), 0, stream>>>(Lr, Li, Pr, Pi, Br, Bi, Cr, Ci, Kre, Kim);
  k_s4    <<<dim3(BATCHN),    dim3(256), LDS_BYTES, stream>>>(x, Mre, Mim, Tre, Tim, Kre, Kim, out);
}